// EquivariantMultiheadAttention_76811195121791
// MI455X (gfx1250) — compile-verified
//
#include <hip/hip_runtime.h>
#include <hip/hip_bf16.h>
#include <math.h>

// ---------------------------------------------------------------------------
// EquivariantMultiheadAttention for MI455X (gfx1250, wave32, WMMA + TDM)
//
// B=4, N=128, S=8, CIN=COUT=8, GDIM=7. pairwise_g = 117 MB fp32 dominates ->
// memory bound (~5us @ 23.3 TB/s). Per (b,i) block: the K=7 dot vs w_g runs
// on v_wmma_f32_16x16x4_f32 (two K=4 steps; B's K=7 row is zero so A's K=7
// slot may be garbage -> packed stride-7 LDS rows are fine). All additive
// logit terms are pre-scaled by log2(e) so softmax exp is a bare v_exp_f32.
// The query term cancels exactly in att/att.sum and is dropped. g tiles are
// staged global->LDS by the Tensor Data Mover (tensor_load_to_lds, TENSORcnt),
// one descriptor per (wave,j), double-buffered.
// ---------------------------------------------------------------------------

typedef float v2f __attribute__((ext_vector_type(2)));
typedef float v8f __attribute__((ext_vector_type(8)));
typedef unsigned int u32x4 __attribute__((ext_vector_type(4)));
typedef unsigned int u32x8 __attribute__((ext_vector_type(8)));

#define Bm    4
#define Nn    128
#define Ss    8
#define Cc    8
#define Gg    7
#define NWAV  8
#define ROWB  (Ss * Ss * Gg)          // 448 floats of g per (i,j)
#define LOG2E 1.44269504088896340736f

#if __has_builtin(__builtin_amdgcn_exp2f)
#define EXP2F __builtin_amdgcn_exp2f
#else
#define EXP2F exp2f
#endif

// ---- Tensor Data Mover: one D# moves 448 contiguous dwords global->LDS ----
// D# group0: {count=1 | lds_addr | global_addr[56:0] | type=2}
// D# group1: {mask=0,data_size=4B | tensor_dim0=448 | tensor_dim1=1 |
//             tile_dim0=448 | tile_dim1=1 | strides=448}
__device__ __forceinline__ void tdm_load_448(unsigned lds_off, const float* gaddr) {
    unsigned long long ga = (unsigned long long)(uintptr_t)gaddr;
    u32x4 g0;
    g0[0] = 1u;                                           // count=1, user D#
    g0[1] = lds_off;                                      // LDS byte address
    g0[2] = (unsigned)ga;                                 // global_addr[31:0]
    g0[3] = (unsigned)((ga >> 32) & 0x01FFFFFFull)        // global_addr[56:32]
          | 0x80000000u;                                  // type=2 ("image")
    u32x8 g1;
    g1[0] = 0x00020000u;                                  // data_size=4B
    g1[1] = (ROWB & 0xFFFFu) << 16;                       // tensor_dim0 lo16
    g1[2] = (ROWB >> 16) | (1u << 16);                    // dim0 hi | dim1 lo
    g1[3] = (ROWB & 0xFFFFu) << 16;                       // dim1 hi | tile_dim0
    g1[4] = 1u;                                           // tile_dim1=1, dim2=0
    g1[5] = (unsigned)ROWB;                               // dim0_stride lo32
    g1[6] = (ROWB & 0xFFFFu) << 16;                       // stride0 hi | stride1 lo
    g1[7] = 0u;                                           // stride1 hi
    asm volatile("tensor_load_to_lds %0, %1" :: "s"(g0), "s"(g1) : "memory");
}
__device__ __forceinline__ void wait_tensor_le1() {       // current buffer done
#if __has_builtin(__builtin_amdgcn_s_wait_tensorcnt)
    __builtin_amdgcn_s_wait_tensorcnt(1);
#else
    asm volatile("s_wait_tensorcnt 0x1");
#endif
    asm volatile("" ::: "memory");
}
__device__ __forceinline__ void wait_tensor_0() {
#if __has_builtin(__builtin_amdgcn_s_wait_tensorcnt)
    __builtin_amdgcn_s_wait_tensorcnt(0);
#else
    asm volatile("s_wait_tensorcnt 0x0");
#endif
    asm volatile("" ::: "memory");
}

__global__ __launch_bounds__(256) void attn_main_kernel(
    const float* __restrict__ gp,          // (B,N,N,S,S,G)
    const float* __restrict__ y,           // (B,N,S,C)
    const unsigned char* __restrict__ mask,// (B,N,S) bool
    const float* __restrict__ w_y,         // (C,2)
    const float* __restrict__ b_y,         // (C)
    const float* __restrict__ w_g,         // (C,G)
    const float* __restrict__ b_g,         // (C)
    float* __restrict__ outstage)          // (B,N,S,C) intermediate
{
    __shared__ float yT[Nn][Cc][Ss];            // yT[j][c][t]          (32 KB)
    __shared__ float kmask[Nn][Ss];             // 0 / -1e30            ( 4 KB)
    __shared__ float wgp[16][8];                // w_g * log2e, padded  (0.5 KB)
    __shared__ float denL[Ss][Cc];
    __shared__ float numL[Ss][Cc];
    __shared__ float gstage[NWAV][2][ROWB];     // packed stride-7 rows (28 KB)

    const int tid = threadIdx.x;
    const int bi  = blockIdx.x;                 // one block per (b,i)
    const int b   = bi >> 7;
    const int i   = bi & (Nn - 1);

    const int wv   = tid >> 5;
    const int lane = tid & 31;
    const int h    = lane >> 4;                 // lane half
    const int col  = lane & 15;                 // D column = channel (valid<8)
    const int cc   = col & 7;

    const size_t gbase = (size_t)((b * Nn + i) * Nn) * ROWB;

    // kick off TDM for this wave's first j immediately (overlaps init staging)
    tdm_load_448((unsigned)(uintptr_t)&gstage[wv][0][0],
                 gp + gbase + (size_t)wv * ROWB);

    // ---- stage per-batch key data & params (pre-scaled by log2e) ----
    for (int idx = tid; idx < Nn * Ss * Cc; idx += 256) {
        int c = idx & 7, t = (idx >> 3) & 7, j = idx >> 6;
        yT[j][c][t] = y[((b * Nn + j) * Ss + t) * Cc + c];
    }
    for (int idx = tid; idx < Nn * Ss; idx += 256) {
        int t = idx & 7, j = idx >> 3;
        kmask[j][t] = mask[(b * Nn + j) * Ss + t] ? 0.0f : -1e30f;
    }
    for (int idx = tid; idx < 16 * 8; idx += 256) {
        int c = idx >> 3, k = idx & 7;
        wgp[c][k] = (c < Cc && k < Gg) ? w_g[c * Gg + k] * LOG2E : 0.0f;
    }
    for (int idx = tid; idx < Ss * Cc; idx += 256) {
        denL[idx >> 3][idx & 7] = 0.0f;
        numL[idx >> 3][idx & 7] = 0.0f;
    }
    __syncthreads();

    // ---- per-lane WMMA fragment setup ----
    v2f bf0, bf1;                               // B[k][n] = log2e * w_g[n][k]
    bf0[0] = wgp[col][2 * h + 0]; bf0[1] = wgp[col][2 * h + 1];
    bf1[0] = wgp[col][2 * h + 4]; bf1[1] = wgp[col][2 * h + 5];  // k=7 row = 0

    const float wy0  = w_y[cc * 2 + 0] * LOG2E;
    const float bias = (b_y[cc] + b_g[cc]) * LOG2E;

    float dacc[4] = {0.f, 0.f, 0.f, 0.f};       // den partials, s = 2*tile+h
    float nacc[4] = {0.f, 0.f, 0.f, 0.f};

    // ---- main loop: wave owns keys j = wv + 8*it, TDM double-buffered ----
    for (int it = 0; it < Nn / NWAV; ++it) {
        const int j = it * NWAV + wv;

        if (it + 1 < Nn / NWAV) {               // prefetch next j, other buffer
            tdm_load_448((unsigned)(uintptr_t)&gstage[wv][(it + 1) & 1][0],
                         gp + gbase + (size_t)(j + NWAV) * ROWB);
            wait_tensor_le1();                  // in-order: current buf done
        } else {
            wait_tensor_0();
        }

        // per-j, per-channel data: raw y (for num) and masked key bias (log2)
        float yv[8];
        v8f   kbv;                              // WMMA C accumulator
#pragma unroll
        for (int v = 0; v < 8; ++v) yv[v] = yT[j][cc][v];
#pragma unroll
        for (int v = 0; v < 8; ++v)
            kbv[v] = __builtin_fmaf(yv[v], wy0, bias) + kmask[j][v];

        const float* gs = &gstage[wv][it & 1][0];
#pragma unroll
        for (int tile = 0; tile < 4; ++tile) {  // 16 rows: s=2*tile+h, t=0..7
            const float* arow = gs + (tile * 16 + col) * Gg;  // stride 7
            v2f a0, a1;
            a0[0] = arow[2 * h + 0]; a0[1] = arow[2 * h + 1];
            a1[0] = arow[2 * h + 4]; a1[1] = arow[2 * h + 5]; // k=7: garbage*0

            v8f d = __builtin_amdgcn_wmma_f32_16x16x4_f32(
                        false, a0, false, bf0, (short)0, kbv, false, false);
            d     = __builtin_amdgcn_wmma_f32_16x16x4_f32(
                        false, a1, false, bf1, (short)0, d,   false, false);

#pragma unroll
            for (int v = 0; v < 8; ++v) {       // e = 2^logit ; masked -> 0
                float e = EXP2F(d[v]);
                dacc[tile] += e;
                nacc[tile] = __builtin_fmaf(e, yv[v], nacc[tile]);
            }
        }
    }

    // ---- cross-wave reduction: one valid lane per (channel, row-half) ----
    if (col < Cc) {
#pragma unroll
        for (int k = 0; k < 4; ++k) {
            int s = 2 * k + h;
            atomicAdd(&denL[s][col], dacc[k]);
            atomicAdd(&numL[s][col], nacc[k]);
        }
    }
    __syncthreads();

    // ---- finalize: residual + normalize + query mask ----
    if (tid < Ss * Cc) {
        int s = tid >> 3, c = tid & 7;
        float yq = y[((b * Nn + i) * Ss + s) * Cc + c];
        float mq = (kmask[i][s] == 0.0f) ? 1.0f : 0.0f;
        float r  = (yq + numL[s][c] / denL[s][c]) * mq;
        outstage[((b * Nn + i) * Ss + s) * Cc + c] = r;
    }
}

// ---------------------------------------------------------------------------
// Output linear: (B*N*S = 4096) x 8 @ w_lin^T (8x8) via WMMA f32 16x16x4.
// ---------------------------------------------------------------------------
__global__ __launch_bounds__(256) void out_linear_kernel(
    const float* __restrict__ xs,
    const float* __restrict__ w_lin,
    float* __restrict__ out)
{
    const int tid  = threadIdx.x;
    const int wv   = tid >> 5;
    const int lane = tid & 31;
    const int tile = blockIdx.x * NWAV + wv;
    const int h    = lane >> 4;
    const int col  = lane & 15;
    const int oo   = col & 7;

    const float* arow = xs + (size_t)(tile * 16 + col) * Cc;
    v2f a0 = *(const v2f*)(arow + 2 * h);
    v2f a1 = *(const v2f*)(arow + 2 * h + 4);

    const float* brow = w_lin + oo * Cc;        // B[k=c][n=o] = w_lin[o][c]
    v2f b0 = *(const v2f*)(brow + 2 * h);
    v2f b1 = *(const v2f*)(brow + 2 * h + 4);

    v8f cz = {0.f, 0.f, 0.f, 0.f, 0.f, 0.f, 0.f, 0.f};
    v8f d = __builtin_amdgcn_wmma_f32_16x16x4_f32(
                false, a0, false, b0, (short)0, cz, false, false);
    d     = __builtin_amdgcn_wmma_f32_16x16x4_f32(
                false, a1, false, b1, (short)0, d,  false, false);

    if (col < Cc) {
#pragma unroll
        for (int v = 0; v < 8; ++v)
            out[(size_t)(tile * 16 + v + 8 * h) * Cc + col] = d[v];
    }
}

// ---------------------------------------------------------------------------
extern "C" void kernel_launch(void* const* d_in, const int* in_sizes, int n_in,
                              void* d_out, int out_size, void* d_ws, size_t ws_size,
                              hipStream_t stream) {
    const float*         gp    = (const float*)d_in[0];
    const float*         y     = (const float*)d_in[1];
    const unsigned char* mask  = (const unsigned char*)d_in[2];
    const float*         w_y   = (const float*)d_in[3];
    const float*         b_y   = (const float*)d_in[4];
    const float*         w_g   = (const float*)d_in[5];
    const float*         b_g   = (const float*)d_in[6];
    const float*         w_lin = (const float*)d_in[7];

    float* stage = (float*)d_ws;                 // B*N*S*CIN = 32768 floats

    attn_main_kernel<<<Bm * Nn, 256, 0, stream>>>(gp, y, mask, w_y, b_y,
                                                  w_g, b_g, stage);
    out_linear_kernel<<<(Bm * Nn * Ss) / (16 * NWAV), 256, 0, stream>>>(
        stage, w_lin, (float*)d_out);
}